// SHEncoder_88450556494138
// MI455X (gfx1250) — compile-verified
//
#include <hip/hip_runtime.h>

// -----------------------------------------------------------------------------
// SH degree-4 encoder as a tall-skinny GEMM:  out[N,16] = Mono[N,20] @ W[20,16]
// via 5x V_WMMA_F32_16X16X4_F32 (K = 20 monomials in 5 chunks of 4).
// Monomial order (K): 1, x, y, z, xx, yy, zz, xy, yz, xz,
//                     xxx, yyy, zzz, xxy, xxz, xyy, yyz, xzz, yzz, xyz
// A 16x4 f32 layout: lanes 0-15 supply K pair (0,1), lanes 16-31 pair (2,3);
// both halves hold the same 16 points (M rows). B mirrors this for W rows.
// D 16x16 f32 layout: VGPR r, half h -> row r+8h, col = lane&15  => fully
// coalesced per-VGPR b32 stores of the row-major [N,16] output.
// -----------------------------------------------------------------------------

typedef __attribute__((ext_vector_type(2))) float v2f;
typedef __attribute__((ext_vector_type(8))) float v8f;

namespace shc {
constexpr float C0   = 0.28209479177387814f;
constexpr float C1   = 0.48860251190291987f;
constexpr float C2_0 = 1.0925484305920792f;
constexpr float C2_1 = -1.0925484305920792f;
constexpr float C2_2 = 0.31539156525252005f;
constexpr float C2_3 = -1.0925484305920792f;
constexpr float C2_4 = 0.5462742152960396f;
constexpr float C3_0 = -0.5900435899266435f;
constexpr float C3_1 = 2.890611442640554f;
constexpr float C3_2 = -0.4570457994644658f;
constexpr float C3_3 = 0.3731763325901154f;
constexpr float C3_4 = -0.4570457994644658f;
constexpr float C3_5 = 1.445305721320277f;
constexpr float C3_6 = -0.5900435899266435f;
}  // namespace shc

// W[k][j]: coefficient of monomial k in SH output column j (20 x 16, sparse).
__device__ __constant__ float SH_W[20][16] = {
    /* 1   */ {shc::C0, 0, 0, 0, 0, 0, 0, 0, 0, 0, 0, 0, 0, 0, 0, 0},
    /* x   */ {0, 0, 0, -shc::C1, 0, 0, 0, 0, 0, 0, 0, 0, 0, 0, 0, 0},
    /* y   */ {0, -shc::C1, 0, 0, 0, 0, 0, 0, 0, 0, 0, 0, 0, 0, 0, 0},
    /* z   */ {0, 0, shc::C1, 0, 0, 0, 0, 0, 0, 0, 0, 0, 0, 0, 0, 0},
    /* xx  */ {0, 0, 0, 0, 0, 0, -shc::C2_2, 0, shc::C2_4, 0, 0, 0, 0, 0, 0, 0},
    /* yy  */ {0, 0, 0, 0, 0, 0, -shc::C2_2, 0, -shc::C2_4, 0, 0, 0, 0, 0, 0, 0},
    /* zz  */ {0, 0, 0, 0, 0, 0, 2.0f * shc::C2_2, 0, 0, 0, 0, 0, 0, 0, 0, 0},
    /* xy  */ {0, 0, 0, 0, shc::C2_0, 0, 0, 0, 0, 0, 0, 0, 0, 0, 0, 0},
    /* yz  */ {0, 0, 0, 0, 0, shc::C2_1, 0, 0, 0, 0, 0, 0, 0, 0, 0, 0},
    /* xz  */ {0, 0, 0, 0, 0, 0, 0, shc::C2_3, 0, 0, 0, 0, 0, 0, 0, 0},
    /* xxx */ {0, 0, 0, 0, 0, 0, 0, 0, 0, 0, 0, 0, 0, -shc::C3_4, 0, shc::C3_6},
    /* yyy */ {0, 0, 0, 0, 0, 0, 0, 0, 0, -shc::C3_0, 0, -shc::C3_2, 0, 0, 0, 0},
    /* zzz */ {0, 0, 0, 0, 0, 0, 0, 0, 0, 0, 0, 0, 2.0f * shc::C3_3, 0, 0, 0},
    /* xxy */ {0, 0, 0, 0, 0, 0, 0, 0, 0, 3.0f * shc::C3_0, 0, -shc::C3_2, 0, 0, 0, 0},
    /* xxz */ {0, 0, 0, 0, 0, 0, 0, 0, 0, 0, 0, 0, -3.0f * shc::C3_3, 0, shc::C3_5, 0},
    /* xyy */ {0, 0, 0, 0, 0, 0, 0, 0, 0, 0, 0, 0, 0, -shc::C3_4, 0, -3.0f * shc::C3_6},
    /* yyz */ {0, 0, 0, 0, 0, 0, 0, 0, 0, 0, 0, 0, -3.0f * shc::C3_3, 0, -shc::C3_5, 0},
    /* xzz */ {0, 0, 0, 0, 0, 0, 0, 0, 0, 0, 0, 0, 0, 4.0f * shc::C3_4, 0, 0},
    /* yzz */ {0, 0, 0, 0, 0, 0, 0, 0, 0, 0, 0, 4.0f * shc::C3_2, 0, 0, 0, 0},
    /* xyz */ {0, 0, 0, 0, 0, 0, 0, 0, 0, 0, shc::C3_1, 0, 0, 0, 0, 0},
};

struct F3 {  // 12-byte packed point -> single global_load_b96
  float x, y, z;
};

__global__ __launch_bounds__(256) void sh_encode_wmma_kernel(
    const float* __restrict__ in, float* __restrict__ out, int n_points) {
  const int lane = (int)(threadIdx.x & 31u);
  const int half = lane >> 4;      // 0: K pair (0,1); 1: K pair (2,3)
  const bool hi  = half != 0;
  const int p    = lane & 15;      // point-in-tile == D column

  const int wave    = (int)((blockIdx.x * blockDim.x + threadIdx.x) >> 5);
  const int n_tiles = (n_points + 15) >> 4;  // 16 points per wave tile
  if (wave >= n_tiles) return;  // wave-uniform guard: EXEC all-ones for WMMA

  // ---- Constant B fragments: 10 explicit b32 loads, one base + const offsets.
  const int kbase = 2 * half;                         // W row offset per half
  const float* wp = &SH_W[kbase][p];                  // per-lane base address
  const float b00 = wp[0 * 16], b10 = wp[1 * 16];     // chunk 0: rows k, k+1
  const float b01 = wp[4 * 16], b11 = wp[5 * 16];     // chunk 1
  const float b02 = wp[8 * 16], b12 = wp[9 * 16];     // chunk 2
  const float b03 = wp[12 * 16], b13 = wp[13 * 16];   // chunk 3
  const float b04 = wp[16 * 16], b14 = wp[17 * 16];   // chunk 4

  // ---- Load this tile's point (both halves read the same 16 points: b96) ----
  const int base = wave << 4;
  int idx = base + p;
  if (idx >= n_points) idx = n_points - 1;  // tail clamp (N % 16 == 0 here)
  const F3 pt = reinterpret_cast<const F3*>(in)[idx];
  const float x = pt.x * 2.0f - 1.0f;
  const float y = pt.y * 2.0f - 1.0f;
  const float z = pt.z * 2.0f - 1.0f;

  // ---- All 20 monomials as named scalars (no private array => no SROA tree).
  const float xx = x * x, yy = y * y, zz = z * z;
  const float xy = x * y, yz = y * z, xz = x * z;
  const float xxx = xx * x, yyy = yy * y, zzz = zz * z;
  const float xxy = xx * y, xxz = xx * z, xyy = x * yy;
  const float yyz = yy * z, xzz = x * zz, yzz = y * zz;
  const float xyz = xy * z;

  // ---- Per-chunk A fragments: lane half selects its two K values (cndmask).
  v2f a0, a1, a2, a3, a4;
  a0[0] = hi ? y : 1.0f;  a0[1] = hi ? z : x;      // K 0..3:  1, x   | y, z
  a1[0] = hi ? zz : xx;   a1[1] = hi ? xy : yy;    // K 4..7:  xx,yy  | zz,xy
  a2[0] = hi ? xxx : yz;  a2[1] = hi ? yyy : xz;   // K 8..11: yz,xz  | xxx,yyy
  a3[0] = hi ? xxz : zzz; a3[1] = hi ? xyy : xxy;  // K 12..15:zzz,xxy| xxz,xyy
  a4[0] = hi ? yzz : yyz; a4[1] = hi ? xyz : xzz;  // K 16..19:yyz,xzz| yzz,xyz

  // ---- 5x V_WMMA_F32_16X16X4_F32 accumulation over K = 20 ----
  v8f acc = {};
  v2f b;
  b[0] = b00; b[1] = b10;
  acc = __builtin_amdgcn_wmma_f32_16x16x4_f32(false, a0, false, b, (short)0,
                                              acc, false, false);
  b[0] = b01; b[1] = b11;
  acc = __builtin_amdgcn_wmma_f32_16x16x4_f32(false, a1, false, b, (short)0,
                                              acc, false, false);
  b[0] = b02; b[1] = b12;
  acc = __builtin_amdgcn_wmma_f32_16x16x4_f32(false, a2, false, b, (short)0,
                                              acc, false, false);
  b[0] = b03; b[1] = b13;
  acc = __builtin_amdgcn_wmma_f32_16x16x4_f32(false, a3, false, b, (short)0,
                                              acc, false, false);
  b[0] = b04; b[1] = b14;
  acc = __builtin_amdgcn_wmma_f32_16x16x4_f32(false, a4, false, b, (short)0,
                                              acc, false, false);

  // ---- Store D. Full-tile fast path is wave-uniform: 8 plain b32 stores,
  // each covering two contiguous 64B segments (rows r and r+8) -> coalesced.
  float* dst = out + (size_t)base * 16u + (size_t)(half * 8) * 16u + p;
  if (base + 16 <= n_points) {
#pragma unroll
    for (int r = 0; r < 8; ++r) dst[r * 16] = acc[r];
  } else {
#pragma unroll
    for (int r = 0; r < 8; ++r) {
      if (base + half * 8 + r < n_points) dst[r * 16] = acc[r];
    }
  }
}

extern "C" void kernel_launch(void* const* d_in, const int* in_sizes, int n_in,
                              void* d_out, int out_size, void* d_ws, size_t ws_size,
                              hipStream_t stream) {
  const float* in = (const float*)d_in[0];
  float* out      = (float*)d_out;
  const int n_points = in_sizes[0] / 3;          // 4,194,304
  const int n_tiles  = (n_points + 15) >> 4;     // 16 points per wave
  const int waves_per_block = 256 / 32;          // wave32: 8 waves/block
  const int blocks = (n_tiles + waves_per_block - 1) / waves_per_block;
  sh_encode_wmma_kernel<<<blocks, 256, 0, stream>>>(in, out, n_points);
}